// RegularizationCalculatorThetaPerStep_89902255440407
// MI455X (gfx1250) — compile-verified
//
#include <hip/hip_runtime.h>
#include <hip/hip_bf16.h>

// Problem constants (match reference)
#define B_N 4096
#define T_N 200
#define D_N 64
#define P_N 2

typedef __attribute__((ext_vector_type(2))) float v2f;
typedef __attribute__((ext_vector_type(8))) float v8f;

// ---------------------------------------------------------------------------
// Full-wave (32 lane) sum via V_WMMA_F32_16X16X4_F32 ones-trick.
// A: a.x of lane L -> A[L%16][L<16 ? 0 : 2], a.y = 0 -> K=1/3 rows are zero.
// B: all ones (layout-independent: every element is 1.0).
// D[m][n] = acc[m] + acc[m+16]  (same value in every column n).
// Sum the 8 D components, then fold partner lane via shfl_xor(16): every lane
// ends with the sum of all 32 accumulators at full f32 precision.
// EXEC must be all 1s at the call site (callers ensure convergence).
// ---------------------------------------------------------------------------
__device__ __forceinline__ float wave_sum_wmma(float v) {
    v2f a; a.x = v;    a.y = 0.0f;
    v2f b; b.x = 1.0f; b.y = 1.0f;
    v8f c = {};
    c = __builtin_amdgcn_wmma_f32_16x16x4_f32(
            /*neg_a=*/false, a, /*neg_b=*/false, b,
            /*c_mod=*/(short)0, c, /*reuse_a=*/false, /*reuse_b=*/false);
    float s = c[0] + c[1] + c[2] + c[3] + c[4] + c[5] + c[6] + c[7];
    s += __shfl_xor(s, 16, 32);
    return s;
}

// Deterministic 256-thread block reduction (8 waves). Result valid in thread 0.
__device__ __forceinline__ float block_sum(float v, float* lds) {
    float s = wave_sum_wmma(v);
    const int wave = threadIdx.x >> 5;
    if ((threadIdx.x & 31) == 0) lds[wave] = s;
    __syncthreads();
    float tot = 0.0f;
    if (threadIdx.x == 0) {
        #pragma unroll
        for (int i = 0; i < 8; ++i) tot += lds[i];
    }
    __syncthreads();   // allow lds reuse by a subsequent reduction
    return tot;
}

// ---------------------------------------------------------------------------
// Kernel 1: per-individual masked sum of squared one-step-ahead errors.
// One 256-thread block per b. Tile = 16 rows x 64 cols: thread (rRow,q)
// handles row rRow, floats 4q..4q+3. preds is 16B-aligned -> b128 load;
// cov rows start at (t+1)*65+1 (misaligned) -> 4x b32. Trip count is
// block-uniform (full tiles) + one masked tail iteration, so the main loop
// is a scalar-branch loop with no exec-mask churn. Rows t >= len-1 are never
// loaded (mask early-exit halves HBM traffic on average).
// ---------------------------------------------------------------------------
__global__ void __launch_bounds__(256)
step_ahead_kernel(const float* __restrict__ cov_trajs,      // [B, T, D+1]
                  const float* __restrict__ preds,          // [B, T-1, D]
                  const int*   __restrict__ lengths,        // [B]
                  float*       __restrict__ per_ind_out) {  // [B]
    __shared__ float lds[8];
    const int b     = blockIdx.x;
    const int len   = lengths[b];
    const int limit = len - 1;                       // rows with mask==1
    const float* covb  = cov_trajs + (size_t)b * (T_N * (D_N + 1));
    const float* predb = preds     + (size_t)b * ((T_N - 1) * D_N);

    const int rRow = threadIdx.x >> 4;    // 0..15 : row inside 16-row tile
    const int q    = threadIdx.x & 15;    // 0..15 : float4 column index

    float acc = 0.0f;
    const int full = limit >> 4;          // uniform tile count
    int t = rRow;
    for (int it = 0; it < full; ++it, t += 16) {
        const float4 pv = *(const float4*)(predb + t * D_N + 4 * q);
        const float* cr = covb + (t + 1) * (D_N + 1) + 1 + 4 * q;
        const float d0 = cr[0] - pv.x;
        const float d1 = cr[1] - pv.y;
        const float d2 = cr[2] - pv.z;
        const float d3 = cr[3] - pv.w;
        acc = fmaf(d0, d0, acc);
        acc = fmaf(d1, d1, acc);
        acc = fmaf(d2, d2, acc);
        acc = fmaf(d3, d3, acc);
    }
    // tail: up to 15 remaining rows, single masked iteration
    if (t < limit) {
        const float4 pv = *(const float4*)(predb + t * D_N + 4 * q);
        const float* cr = covb + (t + 1) * (D_N + 1) + 1 + 4 * q;
        const float d0 = cr[0] - pv.x;
        const float d1 = cr[1] - pv.y;
        const float d2 = cr[2] - pv.z;
        const float d3 = cr[3] - pv.w;
        acc = fmaf(d0, d0, acc);
        acc = fmaf(d1, d1, acc);
        acc = fmaf(d2, d2, acc);
        acc = fmaf(d3, d3, acc);
    }
    // all lanes reconverged here -> EXEC all 1s for WMMA
    const float tot = block_sum(acc, lds);
    if (threadIdx.x == 0) {
        const float cnt = (float)((limit > 0 ? limit : 1) * D_N);
        per_ind_out[b] = (len > 1) ? (tot / cnt) : 0.0f;
    }
}

// ---------------------------------------------------------------------------
// Kernel 2: theta-drift + global-diff terms over pred_params [B, T, 2].
// Each block owns 8 consecutive individuals; thread t (t<200) loads the
// float2 pair {p=0,p=1} at (b,t) and, for t<199, the pair at (b,t+1).
// float2 is always 8B-aligned here -> b64 loads, zero div/mod in the loop.
// ---------------------------------------------------------------------------
__global__ void __launch_bounds__(256)
theta_terms_kernel(const float* __restrict__ pred_params,   // [B, T, P] flat
                   const float* __restrict__ global_theta,  // [P]
                   float*       __restrict__ drift_part,    // [gridDim.x]
                   float*       __restrict__ glob_part) {   // [gridDim.x]
    __shared__ float lds[8];
    const float inv0 = 1.0f / global_theta[0];
    const float inv1 = 1.0f / global_theta[1];

    const int t = threadIdx.x;            // 0..255; active rows: t < 200
    float accD = 0.0f, accG = 0.0f;

    #pragma unroll
    for (int k = 0; k < 8; ++k) {
        const int b = blockIdx.x * 8 + k;
        const float* row = pred_params + ((size_t)b * T_N + t) * P_N;
        if (t < T_N) {
            const float2 v = *(const float2*)row;
            const float g0 = fmaf(v.x, inv0, -1.0f);
            const float g1 = fmaf(v.y, inv1, -1.0f);
            accG = fmaf(g0, g0, accG);
            accG = fmaf(g1, g1, accG);
            if (t < T_N - 1) {
                const float2 n = *(const float2*)(row + P_N);
                const float d0 = v.x - n.x;
                const float d1 = v.y - n.y;
                accD = fmaf(d0, d0, accD);
                accD = fmaf(d1, d1, accD);
            }
        }
    }
    // converged: two WMMA block reductions
    const float dTot = block_sum(accD, lds);
    const float gTot = block_sum(accG, lds);
    if (threadIdx.x == 0) {
        drift_part[blockIdx.x] = dTot;
        glob_part[blockIdx.x]  = gTot;
    }
}

// ---------------------------------------------------------------------------
// Kernel 3: deterministic final combine (single block).
// ---------------------------------------------------------------------------
__global__ void __launch_bounds__(256)
final_kernel(const float* __restrict__ per_ind,     // [B_N]
             const float* __restrict__ drift_part,  // [nPart]
             const float* __restrict__ glob_part,   // [nPart]
             int nPart,
             float* __restrict__ out) {
    __shared__ float lds[8];
    float a1 = 0.0f, a2 = 0.0f, a3 = 0.0f;
    for (int i = threadIdx.x; i < B_N;   i += blockDim.x) a1 += per_ind[i];
    for (int i = threadIdx.x; i < nPart; i += blockDim.x) a2 += drift_part[i];
    for (int i = threadIdx.x; i < nPart; i += blockDim.x) a3 += glob_part[i];

    const float s1 = block_sum(a1, lds);
    const float s2 = block_sum(a2, lds);
    const float s3 = block_sum(a3, lds);
    if (threadIdx.x == 0) {
        const float step_ahead = s1 * (1.0f / (float)B_N);
        const float drift      = s2 * (1.0f / (float)(B_N * (T_N - 1) * P_N));
        const float glob       = s3 * (1.0f / (float)(B_N * T_N * P_N));
        out[0] = 1.0f * step_ahead + 0.1f * drift + 0.01f * glob;
    }
}

// ---------------------------------------------------------------------------
// Launcher. Input order: global_theta, pred_params, hidden_states (UNUSED),
// step_ahead_cov_preds, cov_trajs, lengths.
// ---------------------------------------------------------------------------
extern "C" void kernel_launch(void* const* d_in, const int* in_sizes, int n_in,
                              void* d_out, int out_size, void* d_ws, size_t ws_size,
                              hipStream_t stream) {
    const float* global_theta = (const float*)d_in[0];
    const float* pred_params  = (const float*)d_in[1];
    // d_in[2] = hidden_states: unused by the reference -> never read.
    const float* preds        = (const float*)d_in[3];
    const float* cov_trajs    = (const float*)d_in[4];
    const int*   lengths      = (const int*)d_in[5];
    float* out = (float*)d_out;

    // Workspace layout (floats): [0,4096) per_ind, [4096,4608) drift partials,
    // [4608,5120) global partials. All entries written before being read.
    float* ws        = (float*)d_ws;
    float* per_ind   = ws;
    float* driftPart = ws + B_N;
    float* globPart  = ws + B_N + 512;
    const int nPart = 512;

    step_ahead_kernel<<<B_N, 256, 0, stream>>>(cov_trajs, preds, lengths, per_ind);
    theta_terms_kernel<<<nPart, 256, 0, stream>>>(pred_params, global_theta,
                                                  driftPart, globPart);
    final_kernel<<<1, 256, 0, stream>>>(per_ind, driftPart, globPart, nPart, out);
}